// GCN3D_20349555048856
// MI455X (gfx1250) — compile-verified
//
#include <hip/hip_runtime.h>
#include <hip/hip_bf16.h>
#include <math.h>

#define SUP 7
#define NB 10
#define BATCH 2
#define V0 4096
#define V1 1024
#define V2 256
#define EPSN 1e-12f

typedef float v2f __attribute__((ext_vector_type(2)));
typedef float v8f __attribute__((ext_vector_type(8)));

__device__ __forceinline__ float silu_f(float x) {
    return x / (1.0f + __expf(-x));
}

// ---------------------------------------------------------------------------
// Column-normalize a (3, ncols) direction matrix: out = d / max(||col||, eps)
// ---------------------------------------------------------------------------
__global__ void colnorm_kernel(const float* __restrict__ d, float* __restrict__ o, int ncols) {
    int c = blockIdx.x * blockDim.x + threadIdx.x;
    if (c >= ncols) return;
    float a = d[c], b = d[ncols + c], cc = d[2 * ncols + c];
    float n = sqrtf(a * a + b * b + cc * cc);
    float inv = 1.0f / fmaxf(n, EPSN);
    o[c] = a * inv;
    o[ncols + c] = b * inv;
    o[2 * ncols + c] = cc * inv;
}

// ---------------------------------------------------------------------------
// KNN: per query point keep the 11 smallest d = qi + qj - 2*dot, drop slot 0
// (mimics top_k(-d, k+1)[..., 1:], stable tie-break by lower index).
// V must be a multiple of blockDim.x (256). One thread per query point.
// ---------------------------------------------------------------------------
__global__ void knn_kernel(const float* __restrict__ pts, int V, int* __restrict__ knn_out) {
    int b = blockIdx.y;
    int i = blockIdx.x * blockDim.x + threadIdx.x;
    const float* P = pts + (size_t)b * V * 3;
    __shared__ float sx[256], sy[256], sz[256];

    float px = P[i * 3 + 0], py = P[i * 3 + 1], pz = P[i * 3 + 2];
    float qi = px * px + py * py + pz * pz;

    float bd[NB + 1];
    int   bi[NB + 1];
#pragma unroll
    for (int t = 0; t < NB + 1; ++t) { bd[t] = 3.4e38f; bi[t] = -1; }

    for (int tile = 0; tile < V; tile += 256) {
        __syncthreads();
        int j = tile + threadIdx.x;
        sx[threadIdx.x] = P[j * 3 + 0];
        sy[threadIdx.x] = P[j * 3 + 1];
        sz[threadIdx.x] = P[j * 3 + 2];
        __syncthreads();
        for (int t = 0; t < 256; ++t) {
            float ox = sx[t], oy = sy[t], oz = sz[t];
            float qj = ox * ox + oy * oy + oz * oz;
            float dot = px * ox + py * oy + pz * oz;
            float d = qi + qj - 2.0f * dot;
            if (d < bd[NB]) {
                bd[NB] = d; bi[NB] = tile + t;
#pragma unroll
                for (int s = NB; s > 0; --s) {
                    if (bd[s] < bd[s - 1]) {
                        float td = bd[s]; bd[s] = bd[s - 1]; bd[s - 1] = td;
                        int   ti = bi[s]; bi[s] = bi[s - 1]; bi[s - 1] = ti;
                    }
                }
            }
        }
    }
#pragma unroll
    for (int n = 0; n < NB; ++n)
        knn_out[((size_t)b * V + i) * NB + n] = bi[n + 1];
}

// ---------------------------------------------------------------------------
// Normalized neighbor directions: dirn[b,i,n,:] = (v[idx]-v[i]) / max(||.||,eps)
// ---------------------------------------------------------------------------
__global__ void dirnorm_kernel(const float* __restrict__ pts, const int* __restrict__ knn,
                               int V, int total, float* __restrict__ dirn) {
    int gid = blockIdx.x * blockDim.x + threadIdx.x;
    if (gid >= total) return;
    int n = gid % NB;
    int i = (gid / NB) % V;
    int b = gid / (NB * V);
    const float* P = pts + (size_t)b * V * 3;
    int j = knn[((size_t)b * V + i) * NB + n];
    float dx = P[j * 3 + 0] - P[i * 3 + 0];
    float dy = P[j * 3 + 1] - P[i * 3 + 1];
    float dz = P[j * 3 + 2] - P[i * 3 + 2];
    float nn = sqrtf(dx * dx + dy * dy + dz * dz);
    float inv = 1.0f / fmaxf(nn, EPSN);
    dirn[(size_t)gid * 3 + 0] = dx * inv;
    dirn[(size_t)gid * 3 + 1] = dy * inv;
    dirn[(size_t)gid * 3 + 2] = dz * inv;
}

// ---------------------------------------------------------------------------
// conv_surface: fm0[b,i,c] = silu( sum_s max_n silu(dirn . dcol[s*128+c]) )
// block = (V,B) grid, 128 threads (one per channel)
// ---------------------------------------------------------------------------
__global__ void conv_surface_kernel(const float* __restrict__ dirn, const float* __restrict__ dn,
                                    float* __restrict__ fm0, int V) {
    int b = blockIdx.y, i = blockIdx.x, c = threadIdx.x;
    __shared__ float sd[NB * 3];
    if (threadIdx.x < NB * 3)
        sd[threadIdx.x] = dirn[((size_t)b * V + i) * NB * 3 + threadIdx.x];
    __syncthreads();
    const int NC = SUP * 128;
    float acc = 0.0f;
#pragma unroll
    for (int s = 0; s < SUP; ++s) {
        int col = s * 128 + c;
        float w0 = dn[col], w1 = dn[NC + col], w2 = dn[2 * NC + col];
        float m = -3.4e38f;
#pragma unroll
        for (int n = 0; n < NB; ++n) {
            float t = silu_f(sd[n * 3] * w0 + sd[n * 3 + 1] * w1 + sd[n * 3 + 2] * w2);
            m = fmaxf(m, t);
        }
        acc += m;
    }
    fm0[((size_t)b * V + i) * 128 + c] = silu_f(acc);
}

// ---------------------------------------------------------------------------
// WMMA fp32 GEMM: out[M,N] = A[M,K] @ W[K,N] + bias[N]
// V_WMMA_F32_16X16X4_F32. 4 waves/block, each wave -> 16x64 output strip.
// Requires K%4==0, M%16==0, N%256==0.
// A 16x4 layout: VGPR0 = K {0 | 2} for lanes {0-15 | 16-31}, VGPR1 = K {1 | 3}.
// D 16x16 layout: VGPR r = row r (lanes 0-15), row r+8 (lanes 16-31).
// ---------------------------------------------------------------------------
__global__ void wmma_gemm_bias_kernel(const float* __restrict__ A, const float* __restrict__ W,
                                      const float* __restrict__ bias, float* __restrict__ out,
                                      int M, int K, int N) {
    int wave = threadIdx.x >> 5;
    int lane = threadIdx.x & 31;
    int lrow = lane & 15;
    int kh = (lane >> 4) << 1;   // 0 or 2: K sub-offset this half-wave
    int mh = (lane >> 4) << 3;   // 0 or 8: row offset in C/D
    int m0 = blockIdx.y * 16;
    int n0 = (blockIdx.x * 4 + wave) * 64;
    (void)M;

    v8f acc[4];
#pragma unroll
    for (int t = 0; t < 4; ++t)
#pragma unroll
        for (int e = 0; e < 8; ++e) acc[t][e] = 0.0f;

    for (int k = 0; k < K; k += 4) {
        v2f av;
        av[0] = A[(size_t)(m0 + lrow) * K + k + kh + 0];
        av[1] = A[(size_t)(m0 + lrow) * K + k + kh + 1];
#pragma unroll
        for (int t = 0; t < 4; ++t) {
            v2f bv;
            bv[0] = W[(size_t)(k + kh + 0) * N + n0 + t * 16 + lrow];
            bv[1] = W[(size_t)(k + kh + 1) * N + n0 + t * 16 + lrow];
            acc[t] = __builtin_amdgcn_wmma_f32_16x16x4_f32(
                false, av, false, bv, (short)0, acc[t], false, false);
        }
    }
#pragma unroll
    for (int t = 0; t < 4; ++t) {
        int col = n0 + t * 16 + lrow;
        float bs = bias[col];
#pragma unroll
        for (int r = 0; r < 8; ++r)
            out[(size_t)(m0 + mh + r) * N + col] = acc[t][r] + bs;
    }
}

// ---------------------------------------------------------------------------
// conv_layer core: out[b,i,c] = feat[b,i,c] +
//    sum_s max_n silu(dirn . dcol[s*outc+c]) * feat[b, knn[n], outc + s*outc + c]
// block = (V,B), threads = outc
// ---------------------------------------------------------------------------
__global__ void conv_layer_kernel(const int* __restrict__ knn, const float* __restrict__ dirn,
                                  const float* __restrict__ dwn, const float* __restrict__ feat,
                                  float* __restrict__ out, int V, int outc) {
    int b = blockIdx.y, i = blockIdx.x, c = threadIdx.x;
    __shared__ float sd[NB * 3];
    __shared__ int   sidx[NB];
    if (threadIdx.x < NB * 3)
        sd[threadIdx.x] = dirn[((size_t)b * V + i) * NB * 3 + threadIdx.x];
    if (threadIdx.x < NB)
        sidx[threadIdx.x] = knn[((size_t)b * V + i) * NB + threadIdx.x];
    __syncthreads();

    int F = outc * (SUP + 1);
    int NC = SUP * outc;
    const float* fb = feat + (size_t)b * V * F;
    float acc = fb[(size_t)i * F + c];  // center
#pragma unroll
    for (int s = 0; s < SUP; ++s) {
        int col = s * outc + c;
        float w0 = dwn[col], w1 = dwn[NC + col], w2 = dwn[2 * NC + col];
        float m = -3.4e38f;
#pragma unroll
        for (int n = 0; n < NB; ++n) {
            float th = silu_f(sd[n * 3] * w0 + sd[n * 3 + 1] * w1 + sd[n * 3 + 2] * w2);
            float sup = fb[(size_t)sidx[n] * F + outc + col];
            m = fmaxf(m, th * sup);
        }
        acc += m;
    }
    out[((size_t)b * V + i) * outc + c] = acc;
}

// ---------------------------------------------------------------------------
// BN statistics over rows (deterministic: one block per channel, tree reduce)
// ---------------------------------------------------------------------------
__global__ void bn_stats_kernel(const float* __restrict__ x, int rows, int C,
                                float* __restrict__ mean, float* __restrict__ var) {
    int c = blockIdx.x;
    float s = 0.0f, s2 = 0.0f;
    for (int r = threadIdx.x; r < rows; r += blockDim.x) {
        float v = x[(size_t)r * C + c];
        s += v; s2 += v * v;
    }
    __shared__ float ss[256], ss2[256];
    ss[threadIdx.x] = s; ss2[threadIdx.x] = s2;
    __syncthreads();
    for (int o = 128; o > 0; o >>= 1) {
        if (threadIdx.x < o) {
            ss[threadIdx.x] += ss[threadIdx.x + o];
            ss2[threadIdx.x] += ss2[threadIdx.x + o];
        }
        __syncthreads();
    }
    if (threadIdx.x == 0) {
        float m = ss[0] / (float)rows;
        mean[c] = m;
        var[c] = ss2[0] / (float)rows - m * m;
    }
}

__global__ void bn_silu_kernel(const float* __restrict__ x, const float* __restrict__ mean,
                               const float* __restrict__ var, const float* __restrict__ g,
                               const float* __restrict__ bb, float* __restrict__ out,
                               size_t total, int C, int do_silu) {
    size_t gid = (size_t)blockIdx.x * blockDim.x + threadIdx.x;
    if (gid >= total) return;
    int c = (int)(gid % C);
    float v = (x[gid] - mean[c]) * rsqrtf(var[c] + 1e-5f) * g[c] + bb[c];
    out[gid] = do_silu ? silu_f(v) : v;
}

// ---------------------------------------------------------------------------
// pooled[b,i,c] = max over first-4 knn of fm[b, knn[n], c]
// ---------------------------------------------------------------------------
__global__ void pool4_kernel(const float* __restrict__ fm, const int* __restrict__ knn,
                             float* __restrict__ out, int V, int C, size_t total) {
    size_t gid = (size_t)blockIdx.x * blockDim.x + threadIdx.x;
    if (gid >= total) return;
    int c = (int)(gid % C);
    int i = (int)((gid / C) % V);
    int b = (int)(gid / ((size_t)C * V));
    const int* kn = knn + ((size_t)b * V + i) * NB;
    float m = -3.4e38f;
#pragma unroll
    for (int n = 0; n < 4; ++n)
        m = fmaxf(m, fm[((size_t)b * V + kn[n]) * C + c]);
    out[gid] = m;
}

// ---------------------------------------------------------------------------
// out[b,j,c] = src[b, perm[j], c]
// ---------------------------------------------------------------------------
__global__ void gather_perm_kernel(const float* __restrict__ src, const int* __restrict__ perm,
                                   float* __restrict__ out, int Vin, int Vout, int C, size_t total) {
    size_t gid = (size_t)blockIdx.x * blockDim.x + threadIdx.x;
    if (gid >= total) return;
    int c = (int)(gid % C);
    int j = (int)((gid / C) % Vout);
    int b = (int)(gid / ((size_t)C * Vout));
    out[gid] = src[((size_t)b * Vin + perm[j]) * C + c];
}

// ---------------------------------------------------------------------------
// nearest: out[b,i] = argmin_j ||tgt_i - src_j||^2 (same formula as reference)
// dynamic LDS = Vs*3 floats (entire source set staged in LDS)
// ---------------------------------------------------------------------------
__global__ void nearest_kernel(const float* __restrict__ tgt, const float* __restrict__ src,
                               int Vt, int Vs, int* __restrict__ out) {
    extern __shared__ float ssrc[];
    int b = blockIdx.y;
    int i = blockIdx.x * blockDim.x + threadIdx.x;
    for (int t = threadIdx.x; t < Vs * 3; t += blockDim.x)
        ssrc[t] = src[(size_t)b * Vs * 3 + t];
    __syncthreads();
    if (i >= Vt) return;
    const float* T = tgt + (size_t)b * Vt * 3;
    float px = T[i * 3], py = T[i * 3 + 1], pz = T[i * 3 + 2];
    float qi = px * px + py * py + pz * pz;
    float best = 3.4e38f;
    int bj = 0;
    for (int j = 0; j < Vs; ++j) {
        float ox = ssrc[j * 3], oy = ssrc[j * 3 + 1], oz = ssrc[j * 3 + 2];
        float qj = ox * ox + oy * oy + oz * oz;
        float d = qi + qj - 2.0f * (px * ox + py * oy + pz * oz);
        if (d < best) { best = d; bj = j; }
    }
    out[(size_t)b * Vt + i] = bj;
}

// ---------------------------------------------------------------------------
// Final concat: [fm0(128) | fm1(128) | fm2[n1](256) | fm3[n1](256) | fm4[n2](512)]
// ---------------------------------------------------------------------------
__global__ void concat_kernel(const float* __restrict__ fm0, const float* __restrict__ fm1,
                              const float* __restrict__ fm2, const float* __restrict__ fm3,
                              const float* __restrict__ fm4, const int* __restrict__ n1,
                              const int* __restrict__ n2, float* __restrict__ out, size_t total) {
    size_t gid = (size_t)blockIdx.x * blockDim.x + threadIdx.x;
    if (gid >= total) return;
    int c = (int)(gid % 1280);
    int i = (int)((gid / 1280) % V0);
    int b = (int)(gid / ((size_t)1280 * V0));
    size_t bi0 = (size_t)b * V0 + i;
    float v;
    if (c < 128)       v = fm0[bi0 * 128 + c];
    else if (c < 256)  v = fm1[bi0 * 128 + (c - 128)];
    else if (c < 512)  v = fm2[((size_t)b * V1 + n1[bi0]) * 256 + (c - 256)];
    else if (c < 768)  v = fm3[((size_t)b * V1 + n1[bi0]) * 256 + (c - 512)];
    else               v = fm4[((size_t)b * V2 + n2[bi0]) * 512 + (c - 768)];
    out[gid] = v;
}

// ===========================================================================
extern "C" void kernel_launch(void* const* d_in, const int* in_sizes, int n_in,
                              void* d_out, int out_size, void* d_ws, size_t ws_size,
                              hipStream_t stream) {
    (void)in_sizes; (void)n_in; (void)out_size; (void)ws_size;

    const float* x    = (const float*)d_in[0];
    const float* d0   = (const float*)d_in[1];
    const float* w1   = (const float*)d_in[2];
    const float* b1   = (const float*)d_in[3];
    const float* dir1 = (const float*)d_in[4];
    const float* g1   = (const float*)d_in[5];
    const float* bb1  = (const float*)d_in[6];
    const float* w2   = (const float*)d_in[7];
    const float* b2   = (const float*)d_in[8];
    const float* dir2 = (const float*)d_in[9];
    const float* g2   = (const float*)d_in[10];
    const float* bb2  = (const float*)d_in[11];
    const float* w3   = (const float*)d_in[12];
    const float* b3   = (const float*)d_in[13];
    const float* dir3 = (const float*)d_in[14];
    const float* g3   = (const float*)d_in[15];
    const float* bb3  = (const float*)d_in[16];
    const float* w4   = (const float*)d_in[17];
    const float* b4   = (const float*)d_in[18];
    const float* dir4 = (const float*)d_in[19];
    const int* perm1  = (const int*)d_in[20];
    const int* perm2  = (const int*)d_in[21];
    float* out = (float*)d_out;

    // ---- workspace bump allocator -----------------------------------------
    char* wsp = (char*)d_ws;
    size_t off = 0;
    auto alloc = [&](size_t bytes) -> void* {
        void* p = wsp + off;
        off += (bytes + 255) & ~(size_t)255;
        return p;
    };
    float* d0n   = (float*)alloc(3 * 896 * 4);
    float* dir1n = (float*)alloc(3 * 896 * 4);
    float* dir2n = (float*)alloc(3 * 1792 * 4);
    float* dir3n = (float*)alloc(3 * 1792 * 4);
    float* dir4n = (float*)alloc(3 * 3584 * 4);
    int*   knn_x  = (int*)alloc((size_t)BATCH * V0 * NB * 4);
    int*   knn_v1 = (int*)alloc((size_t)BATCH * V1 * NB * 4);
    int*   knn_v2 = (int*)alloc((size_t)BATCH * V2 * NB * 4);
    float* dirn_x  = (float*)alloc((size_t)BATCH * V0 * NB * 3 * 4);
    float* dirn_v1 = (float*)alloc((size_t)BATCH * V1 * NB * 3 * 4);
    float* dirn_v2 = (float*)alloc((size_t)BATCH * V2 * NB * 3 * 4);
    float* fm0   = (float*)alloc((size_t)BATCH * V0 * 128 * 4);
    float* feat  = (float*)alloc((size_t)BATCH * V0 * 1024 * 4);   // reused all layers
    float* raw   = (float*)alloc((size_t)BATCH * V0 * 128 * 4);    // reused pre-BN buf
    float* fm1   = (float*)alloc((size_t)BATCH * V0 * 128 * 4);
    float* pool1 = (float*)alloc((size_t)BATCH * V0 * 128 * 4);
    float* v1b   = (float*)alloc((size_t)BATCH * V1 * 3 * 4);
    float* f1    = (float*)alloc((size_t)BATCH * V1 * 128 * 4);
    float* fm2   = (float*)alloc((size_t)BATCH * V1 * 256 * 4);
    float* fm3   = (float*)alloc((size_t)BATCH * V1 * 256 * 4);
    float* pool3 = (float*)alloc((size_t)BATCH * V1 * 256 * 4);
    float* v2b   = (float*)alloc((size_t)BATCH * V2 * 3 * 4);
    float* f2    = (float*)alloc((size_t)BATCH * V2 * 256 * 4);
    float* fm4   = (float*)alloc((size_t)BATCH * V2 * 512 * 4);
    float* meanb = (float*)alloc(512 * 4);
    float* varb  = (float*)alloc(512 * 4);
    int*   n1    = (int*)alloc((size_t)BATCH * V0 * 4);
    int*   n2    = (int*)alloc((size_t)BATCH * V0 * 4);

    // ---- normalize direction matrices -------------------------------------
    colnorm_kernel<<<dim3((896 + 255) / 256), 256, 0, stream>>>(d0, d0n, 896);
    colnorm_kernel<<<dim3((896 + 255) / 256), 256, 0, stream>>>(dir1, dir1n, 896);
    colnorm_kernel<<<dim3((1792 + 255) / 256), 256, 0, stream>>>(dir2, dir2n, 1792);
    colnorm_kernel<<<dim3((1792 + 255) / 256), 256, 0, stream>>>(dir3, dir3n, 1792);
    colnorm_kernel<<<dim3((3584 + 255) / 256), 256, 0, stream>>>(dir4, dir4n, 3584);

    // ---- stage 0: knn(x), dirs, conv_surface ------------------------------
    knn_kernel<<<dim3(V0 / 256, BATCH), 256, 0, stream>>>(x, V0, knn_x);
    {
        int total = BATCH * V0 * NB;
        dirnorm_kernel<<<(total + 255) / 256, 256, 0, stream>>>(x, knn_x, V0, total, dirn_x);
    }
    conv_surface_kernel<<<dim3(V0, BATCH), 128, 0, stream>>>(dirn_x, d0n, fm0, V0);

    // ---- layer 1: feat = fm0 @ w1 + b1 ; conv ; bn+silu ; pool ------------
    wmma_gemm_bias_kernel<<<dim3(1024 / 256, (BATCH * V0) / 16), 128, 0, stream>>>(
        fm0, w1, b1, feat, BATCH * V0, 128, 1024);
    conv_layer_kernel<<<dim3(V0, BATCH), 128, 0, stream>>>(knn_x, dirn_x, dir1n, feat, raw, V0, 128);
    bn_stats_kernel<<<128, 256, 0, stream>>>(raw, BATCH * V0, 128, meanb, varb);
    {
        size_t total = (size_t)BATCH * V0 * 128;
        bn_silu_kernel<<<(unsigned)((total + 255) / 256), 256, 0, stream>>>(
            raw, meanb, varb, g1, bb1, fm1, total, 128, 1);
        pool4_kernel<<<(unsigned)((total + 255) / 256), 256, 0, stream>>>(
            fm1, knn_x, pool1, V0, 128, total);
    }
    // subsample to v1 / f1
    {
        size_t t3 = (size_t)BATCH * V1 * 3;
        gather_perm_kernel<<<(unsigned)((t3 + 255) / 256), 256, 0, stream>>>(
            x, perm1, v1b, V0, V1, 3, t3);
        size_t tf = (size_t)BATCH * V1 * 128;
        gather_perm_kernel<<<(unsigned)((tf + 255) / 256), 256, 0, stream>>>(
            pool1, perm1, f1, V0, V1, 128, tf);
    }

    // ---- stage 1: knn(v1), dirs -------------------------------------------
    knn_kernel<<<dim3(V1 / 256, BATCH), 256, 0, stream>>>(v1b, V1, knn_v1);
    {
        int total = BATCH * V1 * NB;
        dirnorm_kernel<<<(total + 255) / 256, 256, 0, stream>>>(v1b, knn_v1, V1, total, dirn_v1);
    }

    // ---- layer 2 ----------------------------------------------------------
    wmma_gemm_bias_kernel<<<dim3(2048 / 256, (BATCH * V1) / 16), 128, 0, stream>>>(
        f1, w2, b2, feat, BATCH * V1, 128, 2048);
    conv_layer_kernel<<<dim3(V1, BATCH), 256, 0, stream>>>(knn_v1, dirn_v1, dir2n, feat, raw, V1, 256);
    bn_stats_kernel<<<256, 256, 0, stream>>>(raw, BATCH * V1, 256, meanb, varb);
    {
        size_t total = (size_t)BATCH * V1 * 256;
        bn_silu_kernel<<<(unsigned)((total + 255) / 256), 256, 0, stream>>>(
            raw, meanb, varb, g2, bb2, fm2, total, 256, 1);
    }

    // ---- layer 3 ----------------------------------------------------------
    wmma_gemm_bias_kernel<<<dim3(2048 / 256, (BATCH * V1) / 16), 128, 0, stream>>>(
        fm2, w3, b3, feat, BATCH * V1, 256, 2048);
    conv_layer_kernel<<<dim3(V1, BATCH), 256, 0, stream>>>(knn_v1, dirn_v1, dir3n, feat, raw, V1, 256);
    bn_stats_kernel<<<256, 256, 0, stream>>>(raw, BATCH * V1, 256, meanb, varb);
    {
        size_t total = (size_t)BATCH * V1 * 256;
        bn_silu_kernel<<<(unsigned)((total + 255) / 256), 256, 0, stream>>>(
            raw, meanb, varb, g3, bb3, fm3, total, 256, 1);
        pool4_kernel<<<(unsigned)((total + 255) / 256), 256, 0, stream>>>(
            fm3, knn_v1, pool3, V1, 256, total);
    }
    // subsample to v2 / f2
    {
        size_t t3 = (size_t)BATCH * V2 * 3;
        gather_perm_kernel<<<(unsigned)((t3 + 255) / 256), 256, 0, stream>>>(
            v1b, perm2, v2b, V1, V2, 3, t3);
        size_t tf = (size_t)BATCH * V2 * 256;
        gather_perm_kernel<<<(unsigned)((tf + 255) / 256), 256, 0, stream>>>(
            pool3, perm2, f2, V1, V2, 256, tf);
    }

    // ---- stage 2: knn(v2), dirs, layer 4 (no bn / no silu) ----------------
    knn_kernel<<<dim3(V2 / 256, BATCH), 256, 0, stream>>>(v2b, V2, knn_v2);
    {
        int total = BATCH * V2 * NB;
        dirnorm_kernel<<<(total + 255) / 256, 256, 0, stream>>>(v2b, knn_v2, V2, total, dirn_v2);
    }
    wmma_gemm_bias_kernel<<<dim3(4096 / 256, (BATCH * V2) / 16), 128, 0, stream>>>(
        f2, w4, b4, feat, BATCH * V2, 256, 4096);
    conv_layer_kernel<<<dim3(V2, BATCH), 512, 0, stream>>>(knn_v2, dirn_v2, dir4n, feat, fm4, V2, 512);

    // ---- nearest-neighbor upsample indices --------------------------------
    nearest_kernel<<<dim3(V0 / 256, BATCH), 256, V1 * 3 * sizeof(float), stream>>>(x, v1b, V0, V1, n1);
    nearest_kernel<<<dim3(V0 / 256, BATCH), 256, V2 * 3 * sizeof(float), stream>>>(x, v2b, V0, V2, n2);

    // ---- final concat -----------------------------------------------------
    {
        size_t total = (size_t)BATCH * V0 * 1280;
        concat_kernel<<<(unsigned)((total + 255) / 256), 256, 0, stream>>>(
            fm0, fm1, fm2, fm3, fm4, n1, n2, out, total);
    }
}